// GCN_4114578669711
// MI455X (gfx1250) — compile-verified
//
#include <hip/hip_runtime.h>

#define NN 50000
#define EE 800000
#define DH 128
#define NCLS 40

typedef __attribute__((ext_vector_type(2))) float v2f;
typedef __attribute__((ext_vector_type(8))) float v8f;

__device__ __forceinline__ void atomAddF(float* p, float v) {
    unsafeAtomicAdd(p, v);   // hardware global_atomic_add_f32 (no return)
}

// ---------------------------------------------------------------- degree/norm
__global__ void init_deg_kernel(float* __restrict__ deg) {
    int i = blockIdx.x * blockDim.x + threadIdx.x;
    if (i < NN) deg[i] = 1.0f;               // self loop contributes 1
}

__global__ void count_deg_kernel(const long long* __restrict__ dst,
                                 float* __restrict__ deg) {
    int e = blockIdx.x * blockDim.x + threadIdx.x;
    if (e < EE) atomAddF(&deg[(int)dst[e]], 1.0f);
}

__global__ void finalize_dinv_kernel(float* __restrict__ deg) {
    int i = blockIdx.x * blockDim.x + threadIdx.x;
    if (i < NN) deg[i] = rsqrtf(deg[i]);     // deg >= 1 always
}

// ---------------------------------------------------------------- WMMA GEMM
// out[M x Nout] = act(A[M x 128]) @ W[128 x Nout] (+ bias)
// M = 50000 = 3125 * 16 exactly. One block = one 16-row tile; 8 waves cover
// the N tiles. A tile AND transposed zero-padded W staged in LDS so the
// k-loop is pure ds_load_b64 + v_wmma (no predicated global loads, no
// loadcnt stalls). Stride 132 => conflict-free b64 fragment loads.
template<bool RELU_A, bool ADD_BIAS>
__global__ void __launch_bounds__(256)
gemm128_wmma_kernel(const float* __restrict__ A, const float* __restrict__ W,
                    const float* __restrict__ bias, float* __restrict__ out,
                    int Nout) {
    extern __shared__ float smem[];
    float* sA  = smem;               // [16][132]  A tile, row major
    float* sWT = smem + 16 * 132;    // [Np][132]  W transposed (col-major), padded
    const int Np  = (Nout + 15) & ~15;
    const int tid  = threadIdx.x;
    const int wid  = tid >> 5;
    const int lane = tid & 31;
    const int m0   = blockIdx.x * 16;

    // stage 16x128 A tile (512 float4), fuse relu
    const float4* A4 = (const float4*)(A + (size_t)m0 * DH);
    for (int i = tid; i < 512; i += 256) {
        float4 v = A4[i];
        if (RELU_A) {
            v.x = fmaxf(v.x, 0.f); v.y = fmaxf(v.y, 0.f);
            v.z = fmaxf(v.z, 0.f); v.w = fmaxf(v.w, 0.f);
        }
        int r  = i >> 5;
        int c4 = (i & 31) * 4;
        float* d = sA + r * 132 + c4;
        d[0] = v.x; d[1] = v.y; d[2] = v.z; d[3] = v.w;
    }
    // stage W transposed: sWT[n][k] = W[k][n], zero-pad cols [Nout, Np)
    for (int k = wid; k < DH; k += 8) {
        const float* Wk = W + (size_t)k * Nout;
        for (int n = lane; n < Np; n += 32) {
            sWT[n * 132 + k] = (n < Nout) ? Wk[n] : 0.f;
        }
    }
    __syncthreads();

    const int half = lane >> 4;      // 0: lanes 0-15, 1: lanes 16-31
    const int lr   = lane & 15;
    const float* pA = sA + lr * 132 + 2 * half;

    for (int nt = wid; nt < (Np >> 4); nt += 8) {
        const int col = nt * 16 + lr;
        const float* pB = sWT + col * 132 + 2 * half;
        v8f c = {};
        #pragma unroll
        for (int k0 = 0; k0 < DH; k0 += 4) {
            // A 16x4 frag: lanes 0-15 -> K=k0,k0+1 ; lanes 16-31 -> K=k0+2,k0+3
            v2f a = *(const v2f*)(pA + k0);   // one ds_load_b64
            v2f b = *(const v2f*)(pB + k0);   // one ds_load_b64
            c = __builtin_amdgcn_wmma_f32_16x16x4_f32(
                    false, a, false, b, (short)0, c, false, false);
        }
        if (col < Nout) {
            float bval = ADD_BIAS ? bias[col] : 0.f;
            #pragma unroll
            for (int i = 0; i < 8; ++i) {
                int r = m0 + i + half * 8;   // C layout: VGPR i -> M=i / M=i+8
                out[(size_t)r * Nout + col] = c[i] + bval;
            }
        }
    }
}

// ---------------------------------------------------------------- aggregation
// acc[i,d] = h[i,d] * dinv[i]^2 (self loop) + bias[d]
__global__ void init_acc_kernel(const float* __restrict__ h,
                                const float* __restrict__ dinv,
                                const float* __restrict__ bias,
                                float* __restrict__ acc) {
    int t = blockIdx.x * blockDim.x + threadIdx.x;
    if (t >= NN * DH) return;
    int node = t >> 7;
    int d    = t & (DH - 1);
    float dv = dinv[node];
    acc[t] = h[t] * dv * dv + bias[d];
}

// one wave per edge: coalesced 128-float gather of h[src], scaled scatter-add
__global__ void scatter_edges_kernel(const long long* __restrict__ src,
                                     const long long* __restrict__ dst,
                                     const float* __restrict__ dinv,
                                     const float* __restrict__ h,
                                     float* __restrict__ acc) {
    int t = blockIdx.x * blockDim.x + threadIdx.x;
    int e = t >> 5;
    if (e >= EE) return;
    int lane = t & 31;
    int s = (int)src[e];
    int d = (int)dst[e];
    float nrm = dinv[s] * dinv[d];
    float4 hv = ((const float4*)(h + (size_t)s * DH))[lane];
    float* ap = acc + (size_t)d * DH + lane * 4;
    atomAddF(ap + 0, hv.x * nrm);
    atomAddF(ap + 1, hv.y * nrm);
    atomAddF(ap + 2, hv.z * nrm);
    atomAddF(ap + 3, hv.w * nrm);
}

// ---------------------------------------------------------------- launcher
extern "C" void kernel_launch(void* const* d_in, const int* in_sizes, int n_in,
                              void* d_out, int out_size, void* d_ws, size_t ws_size,
                              hipStream_t stream) {
    const float*     x   = (const float*)d_in[0];
    const long long* ei  = (const long long*)d_in[1];   // int64 [2, E]
    const float*     W1  = (const float*)d_in[2];
    const float*     b1  = (const float*)d_in[3];
    const float*     Wh  = (const float*)d_in[4];       // [2,128,128]
    const float*     bh  = (const float*)d_in[5];       // [2,128]
    const float*     Wf  = (const float*)d_in[6];       // [128,40]
    const float*     bf  = (const float*)d_in[7];       // [40]
    float*           out = (float*)d_out;               // [50000,40]

    const long long* srcI = ei;
    const long long* dstI = ei + EE;

    float* ws   = (float*)d_ws;
    float* dinv = ws;                              // 50000 floats
    float* bufA = ws + 50048;                      // N*128
    float* bufB = bufA + (size_t)NN * DH;          // N*128

    const int nodeBlks = (NN + 255) / 256;
    const int edgeBlks = (EE + 255) / 256;
    const int elemBlks = (NN * DH) / 256;          // 25000
    const int scatBlks = (EE * 32) / 256;          // 100000
    const int gemmBlks = NN / 16;                  // 3125 (exact)

    const size_t shmemHidden = (size_t)(16 * 132 + 128 * 132) * sizeof(float); // ~74 KB
    const size_t shmemHead   = (size_t)(16 * 132 +  48 * 132) * sizeof(float); // ~33 KB

    // normalization
    init_deg_kernel<<<nodeBlks, 256, 0, stream>>>(dinv);
    count_deg_kernel<<<edgeBlks, 256, 0, stream>>>(dstI, dinv);
    finalize_dinv_kernel<<<nodeBlks, 256, 0, stream>>>(dinv);

    // layer 1: x -> bufA (transform), bufB (aggregated pre-relu)
    gemm128_wmma_kernel<false, false><<<gemmBlks, 256, shmemHidden, stream>>>(x, W1, nullptr, bufA, DH);
    init_acc_kernel<<<elemBlks, 256, 0, stream>>>(bufA, dinv, b1, bufB);
    scatter_edges_kernel<<<scatBlks, 256, 0, stream>>>(srcI, dstI, dinv, bufA, bufB);

    // hidden layers: relu fused into GEMM A-read
    for (int l = 0; l < 2; ++l) {
        const float* Wl = Wh + (size_t)l * DH * DH;
        const float* bl = bh + (size_t)l * DH;
        gemm128_wmma_kernel<true, false><<<gemmBlks, 256, shmemHidden, stream>>>(bufB, Wl, nullptr, bufA, DH);
        init_acc_kernel<<<elemBlks, 256, 0, stream>>>(bufA, dinv, bl, bufB);
        scatter_edges_kernel<<<scatBlks, 256, 0, stream>>>(srcI, dstI, dinv, bufA, bufB);
    }

    // head: relu(bufB) @ Wf + bf -> out  (Nout = 40)
    gemm128_wmma_kernel<true, true><<<gemmBlks, 256, shmemHead, stream>>>(bufB, Wf, bf, out, NCLS);
}